// SchmidtQuantumCircuit_28037546508635
// MI455X (gfx1250) — compile-verified
//
#include <hip/hip_runtime.h>
#include <math.h>

// CDNA5 / gfx1250, wave32. Matrix ops via v_wmma_f32_16x16x4_f32.

typedef __attribute__((ext_vector_type(2))) float v2f;
typedef __attribute__((ext_vector_type(8))) float v8f;

#define NQ      18
#define NANC    4
#define AA      16          // 2^NANC
#define NLAYERS 2
#define BATCH   8
#define DDIM    (1 << 18)
#define RDIM    512         // 2^9 (matrix side)

// Workspace layout (floats)
#define WS_WC 0             // [18] sum_a anc^2 cos(theta_aq)
#define WS_WS 18            // [18] sum_a anc^2 sin(theta_aq)
#define WS_S2 36            // [8]  ||x_b||^2
#define WS_Z  44            // [8][18] sign-weighted quadratic sums (by bit position p)
#define WS_X  (44 + 144)    // [8][18] xor-pair sums (by bit position p)
#define WS_TOTAL (44 + 288)

// ---------------------------------------------------------------------------
// Kernel 1: simulate 4-qubit ancilla circuit, fold into per-qubit weights.
// Tiny (16-vector, 14 gates) -> one lane does it all.
// ---------------------------------------------------------------------------
__global__ void anc_weights_kernel(const float* __restrict__ ap,
                                   const float* __restrict__ up,
                                   float* __restrict__ ws) {
  if (threadIdx.x != 0) return;
  float vec[AA];
  for (int i = 0; i < AA; ++i) vec[i] = 0.0f;
  vec[0] = 1.0f;
  int p = 0;
  for (int layer = 0; layer < NLAYERS; ++layer) {
    for (int w = 0; w < NANC; ++w) {            // RY on wire w (bit pos NANC-1-w)
      float th = ap[p++];
      float c = cosf(0.5f * th), s = sinf(0.5f * th);
      int stride = 1 << (NANC - 1 - w);
      for (int i = 0; i < AA; ++i) {
        if (i & stride) continue;
        float v0 = vec[i], v1 = vec[i + stride];
        vec[i]          = c * v0 - s * v1;
        vec[i + stride] = s * v0 + c * v1;
      }
    }
    for (int w = 0; w < NANC - 1; ++w) {        // CRY ctrl=w, tgt=w+1
      float th = ap[p++];
      float c = cosf(0.5f * th), s = sinf(0.5f * th);
      int cb = 1 << (NANC - 1 - w);
      int tb = 1 << (NANC - 1 - (w + 1));
      for (int i = 0; i < AA; ++i) {
        if (!(i & cb)) continue;
        if (i & tb) continue;
        float v0 = vec[i], v1 = vec[i + tb];
        vec[i]      = c * v0 - s * v1;
        vec[i + tb] = s * v0 + c * v1;
      }
    }
  }
  for (int q = 0; q < NQ; ++q) {
    float wc = 0.0f, wsn = 0.0f;
    for (int a = 0; a < AA; ++a) {
      float w2 = vec[a] * vec[a];
      float th = up[a * NQ + q];
      wc  += w2 * cosf(th);   // cos(theta) = c^2 - s^2
      wsn += w2 * sinf(th);   // sin(theta) = 2 c s
    }
    ws[WS_WC + q] = wc;
    ws[WS_WS + q] = wsn;
  }
}

// ---------------------------------------------------------------------------
// Kernel 2: one wave per needed 16x16 Gram tile, templated on strides so the
// contiguous-K case (G = M M^T, CS==1) lowers to global_load_b64 and constant
// instruction offsets, while the strided case (H = M^T M, CS==512) keeps b32.
//   RS=512,CS=1,BASE=9 : G over hi index -> bits p = 9..17
//   RS=1,CS=512,BASE=0 : H over lo index -> bits p = 0..8
//   jsel 0: diagonal tile  (diag -> S2/Z, in-tile xor diags j=0..3 -> X)
//   jsel 1..5: J = I ^ 2^(jsel-1), tile diagonal -> X for j = 4..8
// K = 512 accumulated with 128x v_wmma_f32_16x16x4_f32 (exact f32 path),
// split into two independent accumulator chains for back-to-back XDL issue.
// ---------------------------------------------------------------------------
template <int RS, int CS, int BASE>
__global__ void gram_tile_kernel(const float* __restrict__ x,
                                 float* __restrict__ ws) {
  const int lane = threadIdx.x;            // 0..31, all active (EXEC all ones)
  int t = blockIdx.x;
  const int jsel = t % 6;  t /= 6;
  const int I    = t % 32; t /= 32;
  const int b    = t;                      // 0..7
  const int J    = (jsel == 0) ? I : (I ^ (1 << (jsel - 1)));

  const float* xb = x + (size_t)b * DDIM;
  const int m    = lane & 15;
  const int koff = (lane >> 4) * 2;        // lanes 16-31 carry K+2,K+3 (ISA A/B layout)
  const float* aRow = xb + (size_t)(16 * I + m) * RS;
  const float* bRow = xb + (size_t)(16 * J + m) * RS;

  v8f acc0 = {};
  v8f acc1 = {};
#pragma unroll 8
  for (int k = 0; k < RDIM; k += 8) {      // two K=4 steps per iteration
    v2f a0, b0, a1, b1;
    if (CS == 1) {
      // 8-byte aligned (koff even, rows 512-float aligned) -> global_load_b64
      float2 ta0 = *reinterpret_cast<const float2*>(aRow + k + koff);
      float2 tb0 = *reinterpret_cast<const float2*>(bRow + k + koff);
      float2 ta1 = *reinterpret_cast<const float2*>(aRow + k + 4 + koff);
      float2 tb1 = *reinterpret_cast<const float2*>(bRow + k + 4 + koff);
      a0.x = ta0.x; a0.y = ta0.y;  b0.x = tb0.x; b0.y = tb0.y;
      a1.x = ta1.x; a1.y = ta1.y;  b1.x = tb1.x; b1.y = tb1.y;
    } else {
      a0.x = aRow[(size_t)(k + koff)         * CS];
      a0.y = aRow[(size_t)(k + koff + 1)     * CS];
      b0.x = bRow[(size_t)(k + koff)         * CS];
      b0.y = bRow[(size_t)(k + koff + 1)     * CS];
      a1.x = aRow[(size_t)(k + 4 + koff)     * CS];
      a1.y = aRow[(size_t)(k + 4 + koff + 1) * CS];
      b1.x = bRow[(size_t)(k + 4 + koff)     * CS];
      b1.y = bRow[(size_t)(k + 4 + koff + 1) * CS];
    }
    // D = A(16x4) * B(4x16) + C ; 8 args: neg_a, A, neg_b, B, c_mod, C, reuse_a, reuse_b
    acc0 = __builtin_amdgcn_wmma_f32_16x16x4_f32(false, a0, false, b0,
                                                 (short)0, acc0, false, false);
    acc1 = __builtin_amdgcn_wmma_f32_16x16x4_f32(false, a1, false, b1,
                                                 (short)0, acc1, false, false);
  }

  // Spill D tile to LDS per ISA C/D layout: VGPR r of lane L holds
  // D[M = r + 8*(L>>4)][N = L&15].
  __shared__ float Dl[16 * 16];
#pragma unroll
  for (int r = 0; r < 8; ++r) {
    int M = r + 8 * (lane >> 4);
    Dl[M * 16 + m] = acc0[r] + acc1[r];
  }
  __syncthreads();

  if (lane == 0) {
    if (jsel == 0) {
      float s2 = 0.0f;
      float zs[9], xs[4];
      for (int j = 0; j < 9; ++j) zs[j] = 0.0f;
      for (int j = 0; j < 4; ++j) xs[j] = 0.0f;
      for (int r = 0; r < 16; ++r) {
        int i = 16 * I + r;                // gram-local 9-bit index
        float dv = Dl[r * 16 + r];
        s2 += dv;
        for (int j = 0; j < 9; ++j) zs[j] += ((i >> j) & 1) ? -dv : dv;
        for (int j = 0; j < 4; ++j) xs[j] += Dl[r * 16 + (r ^ (1 << j))];
      }
      if (BASE == 9) atomicAdd(&ws[WS_S2 + b], s2);  // trace(G) = ||x||^2 (count once)
      for (int j = 0; j < 9; ++j) atomicAdd(&ws[WS_Z + b * NQ + BASE + j], zs[j]);
      for (int j = 0; j < 4; ++j) atomicAdd(&ws[WS_X + b * NQ + BASE + j], xs[j]);
    } else {
      float xsum = 0.0f;
      for (int r = 0; r < 16; ++r) xsum += Dl[r * 16 + r];
      atomicAdd(&ws[WS_X + b * NQ + BASE + 4 + (jsel - 1)], xsum);
    }
  }
}

// ---------------------------------------------------------------------------
// Kernel 3: out[b,q] = (Wc[q]*Z[b,p] - Ws[q]*X[b,p]) / ||x_b||^2 ,  p = 17-q
// ---------------------------------------------------------------------------
__global__ void finalize_kernel(const float* __restrict__ ws,
                                float* __restrict__ out) {
  int t = blockIdx.x * blockDim.x + threadIdx.x;
  if (t >= BATCH * NQ) return;
  int b = t / NQ, q = t % NQ;
  int p = (NQ - 1) - q;
  float s2 = ws[WS_S2 + b];
  float z  = ws[WS_Z + b * NQ + p];
  float xv = ws[WS_X + b * NQ + p];
  out[b * NQ + q] = (ws[WS_WC + q] * z - ws[WS_WS + q] * xv) / s2;
}

extern "C" void kernel_launch(void* const* d_in, const int* in_sizes, int n_in,
                              void* d_out, int out_size, void* d_ws, size_t ws_size,
                              hipStream_t stream) {
  const float* x  = (const float*)d_in[0];   // (8, 2^18)
  const float* ap = (const float*)d_in[1];   // (14,)
  const float* up = (const float*)d_in[2];   // (16, 18)
  float* ws  = (float*)d_ws;
  float* out = (float*)d_out;                // (8, 18)

  hipMemsetAsync(d_ws, 0, WS_TOTAL * sizeof(float), stream);
  anc_weights_kernel<<<1, 32, 0, stream>>>(ap, up, ws);
  // G = M M^T (hi bits, contiguous K)   and   H = M^T M (lo bits, strided K)
  gram_tile_kernel<RDIM, 1, 9><<<BATCH * 32 * 6, 32, 0, stream>>>(x, ws);
  gram_tile_kernel<1, RDIM, 0><<<BATCH * 32 * 6, 32, 0, stream>>>(x, ws);
  finalize_kernel<<<1, 192, 0, stream>>>(ws, out);
}